// UpConvLayers_30683246363153
// MI455X (gfx1250) — compile-verified
//
#include <hip/hip_runtime.h>

typedef __attribute__((ext_vector_type(2))) float v2f;
typedef __attribute__((ext_vector_type(8))) float v8f;

#define N_NODES 100000

// ---------------- utility kernels ----------------

__global__ void zero_kernel(float* __restrict__ p, long long n) {
    long long i = (long long)blockIdx.x * blockDim.x + threadIdx.x;
    if (i < n) p[i] = 0.0f;
}

// deg accumulation: one atomicAdd per edge into deg[dst]
__global__ void degree_kernel(const long long* __restrict__ dst,
                              float* __restrict__ deg, int E) {
    int i = blockIdx.x * blockDim.x + threadIdx.x;
    if (i < E) atomicAdd(&deg[(int)dst[i]], 1.0f);
}

// dis[i] = rsqrt(deg[i] + 1)   (the +1 is the self-loop; deg >= 1 always)
__global__ void dis_kernel(float* __restrict__ d, int n) {
    int i = blockIdx.x * blockDim.x + threadIdx.x;
    if (i < n) d[i] = rsqrtf(d[i] + 1.0f);
}

// ---------------- WMMA fp32 GEMM: H[N,Fout] = A[N,Fin] @ W[Fin,Fout] ----------------
// One wave computes one 16x16 tile of H, accumulating with v_wmma_f32_16x16x4_f32.
// A-frag (16x4 fp32, 2 VGPR): lanes 0-15 -> {K=k0,k0+1}, lanes 16-31 -> {K=k0+2,k0+3}, M=lane&15
// B-frag (4x16 fp32, 2 VGPR): mirrored, N=lane&15
// C/D (16x16 fp32, 8 VGPR):   vgpr v -> M = v + 8*(lane>>4), N = lane&15
__global__ __launch_bounds__(256)
void gemm_wmma_f32(const float* __restrict__ A, const float* __restrict__ W,
                   float* __restrict__ H, int Fin, int Fout, int Mtiles) {
    const int lane  = threadIdx.x & 31;
    const int wave  = threadIdx.x >> 5;
    const int mtile = blockIdx.y * 8 + wave;
    if (mtile >= Mtiles) return;               // uniform per wave; EXEC stays all-ones
    const int ntile = blockIdx.x;

    const int mloc = lane & 15;
    const int kb   = (lane >> 4) << 1;         // 0 for lanes 0-15, 2 for lanes 16-31

    const float* arow = A + (size_t)(mtile * 16 + mloc) * Fin + kb;
    const float* wcol = W + (size_t)kb * Fout + (ntile * 16 + mloc);

    v8f acc = {};
    for (int k0 = 0; k0 < Fin; k0 += 4) {
        v2f a = *(const v2f*)(arow + k0);                // A[m][k0+kb], A[m][k0+kb+1]
        v2f b;
        b.x = wcol[(size_t)k0 * Fout];                   // W[k0+kb  ][n]
        b.y = wcol[(size_t)(k0 + 1) * Fout];             // W[k0+kb+1][n]
        acc = __builtin_amdgcn_wmma_f32_16x16x4_f32(
            /*neg_a=*/false, a, /*neg_b=*/false, b,
            /*c_mod=*/(short)0, acc, /*reuse_a=*/false, /*reuse_b=*/false);
    }

    const int mrow0 = (lane >> 4) << 3;                  // 0 or 8
    float* hout = H + (size_t)(mtile * 16) * Fout + ntile * 16 + mloc;
#pragma unroll
    for (int v = 0; v < 8; ++v)
        hout[(size_t)(mrow0 + v) * Fout] = acc[v];
}

// ---------------- edge scatter: agg[dst] += dis[src]*dis[dst] * h[src] ----------------
// One block per edge, F/4 threads, float4 gather (L2-resident) + 4 atomic adds.
__global__ void scatter_kernel(const float* __restrict__ H,
                               const long long* __restrict__ src,
                               const long long* __restrict__ dst,
                               const float* __restrict__ dis,
                               float* __restrict__ agg, int F) {
    const int e = blockIdx.x;
    const int s = (int)src[e];
    const int d = (int)dst[e];
    const float nrm = dis[s] * dis[d];
    const float4 hv = ((const float4*)(H + (size_t)s * F))[threadIdx.x];
    float* ap = agg + (size_t)d * F + threadIdx.x * 4;
    atomicAdd(ap + 0, nrm * hv.x);
    atomicAdd(ap + 1, nrm * hv.y);
    atomicAdd(ap + 2, nrm * hv.z);
    atomicAdd(ap + 3, nrm * hv.w);
}

// ---------------- fuse: out = [relu](agg + dis^2 * h + b)  (self-loop folded in) ----------------
__global__ void fuse_kernel(const float* agg, const float* __restrict__ H,
                            const float* __restrict__ dis,
                            const float* __restrict__ bias,
                            float* out, int F, int do_relu) {
    const int i = blockIdx.x;
    const int f = threadIdx.x;
    const float di = dis[i];
    float v = agg[(size_t)i * F + f] + di * di * H[(size_t)i * F + f] + bias[f];
    if (do_relu) v = fmaxf(v, 0.0f);
    out[(size_t)i * F + f] = v;
}

// ---------------- launcher ----------------

extern "C" void kernel_launch(void* const* d_in, const int* in_sizes, int n_in,
                              void* d_out, int out_size, void* d_ws, size_t ws_size,
                              hipStream_t stream) {
    const float*     x   = (const float*)d_in[0];
    const long long* ei  = (const long long*)d_in[1];
    const float*     W3  = (const float*)d_in[2];
    const float*     b3  = (const float*)d_in[3];
    const float*     W4  = (const float*)d_in[4];
    const float*     b4  = (const float*)d_in[5];
    const float*     W5  = (const float*)d_in[6];
    const float*     b5  = (const float*)d_in[7];

    const int N = N_NODES;
    const int E = in_sizes[1] / 2;
    const long long* src = ei;          // edge_index[0]
    const long long* dst = ei + E;      // edge_index[1]

    // workspace layout: dis[N] | h[N*256] | agg[N*256]
    float* dis = (float*)d_ws;
    float* h   = dis + N;
    float* agg = h + (size_t)N * 256;

    const int Mtiles = N / 16;          // 6250

    // --- degrees -> deg_inv_sqrt ---
    zero_kernel<<<(N + 255) / 256, 256, 0, stream>>>(dis, (long long)N);
    degree_kernel<<<(E + 255) / 256, 256, 0, stream>>>(dst, dis, E);
    dis_kernel<<<(N + 255) / 256, 256, 0, stream>>>(dis, N);

    // --- layer 3: Fin=64, Fout=128, relu ---
    {
        const int Fin = 64, Fout = 128;
        dim3 grid(Fout / 16, (Mtiles + 7) / 8);
        gemm_wmma_f32<<<grid, 256, 0, stream>>>(x, W3, h, Fin, Fout, Mtiles);
        long long na = (long long)N * Fout;
        zero_kernel<<<(unsigned)((na + 255) / 256), 256, 0, stream>>>(agg, na);
        scatter_kernel<<<E, Fout / 4, 0, stream>>>(h, src, dst, dis, agg, Fout);
        fuse_kernel<<<N, Fout, 0, stream>>>(agg, h, dis, b3, agg, Fout, 1);
    }

    // --- layer 4: Fin=128, Fout=256, relu ---
    {
        const int Fin = 128, Fout = 256;
        dim3 grid(Fout / 16, (Mtiles + 7) / 8);
        gemm_wmma_f32<<<grid, 256, 0, stream>>>(agg, W4, h, Fin, Fout, Mtiles);
        long long na = (long long)N * Fout;
        zero_kernel<<<(unsigned)((na + 255) / 256), 256, 0, stream>>>(agg, na);
        scatter_kernel<<<E, Fout / 4, 0, stream>>>(h, src, dst, dis, agg, Fout);
        fuse_kernel<<<N, Fout, 0, stream>>>(agg, h, dis, b4, agg, Fout, 1);
    }

    // --- layer 5: Fin=256, Fout=128, no relu, write d_out ---
    {
        const int Fin = 256, Fout = 128;
        dim3 grid(Fout / 16, (Mtiles + 7) / 8);
        gemm_wmma_f32<<<grid, 256, 0, stream>>>(agg, W5, h, Fin, Fout, Mtiles);
        long long na = (long long)N * Fout;
        zero_kernel<<<(unsigned)((na + 255) / 256), 256, 0, stream>>>(agg, na);
        scatter_kernel<<<E, Fout / 4, 0, stream>>>(h, src, dst, dis, agg, Fout);
        fuse_kernel<<<N, Fout, 0, stream>>>(agg, h, dis, b5, (float*)d_out, Fout, 0);
    }
}